// EGNNLayer_16458314678570
// MI455X (gfx1250) — compile-verified
//
#include <hip/hip_runtime.h>
#include <math.h>

// ---------------------------------------------------------------------------
// EGNN layer for MI455X (gfx1250, wave32).
//
// v2 changes (driven by round-1 asm):
//  * 2 row-tiles x 4 col-tiles register blocking -> 8 wmma per 6 loads
//    (was 1 wmma per ~2.3 loads with a full s_wait between each).
//  * Weights pre-packed into pair-interleaved layout so each B fragment
//    ({W[k][col], W[k+1][col]}) is ONE 8-byte coalesced load instead of two
//    512B-strided dword loads.
//  * 32 edges per wave: every lane owns one edge for geometry/gate work.
//
// WMMA operand layouts (ISA 7.12.2, wave32), V_WMMA_F32_16X16X4_F32:
//  A 16x4 f32 : lane L -> row L%16, K = 2*(L/16) + v      (v2f per lane)
//  B 4x16 f32 : lane L -> col L%16, K = v + 2*(L/16)
//  C/D 16x16  : vgpr r -> row r + 8*(L/16), col L%16      (v8f per lane)
// ---------------------------------------------------------------------------

#define N_NODES 50000
#define N_EDGES 800000
#define HDIM    128
#define EDIM    16
#define LDS_LD  132        // padded row stride (floats): bank-conflict free

typedef float v2f __attribute__((ext_vector_type(2)));
typedef float v8f __attribute__((ext_vector_type(8)));

__device__ __forceinline__ v8f wmma_f32(v2f a, v2f b, v8f c) {
    return __builtin_amdgcn_wmma_f32_16x16x4_f32(
        false, a, false, b, (short)0, c, false, false);
}

__device__ __forceinline__ float silu_f(float v) {
    return v / (1.0f + __expf(-v));
}

// ---------------------------------------------------------------------------
// K-loop, 2 row-tiles x 4 col-tiles.  A0/A1: this lane's A-row pointers for
// the two row tiles.  Wp: pair-packed weights (v2f element = rows {2p,2p+1}
// at one column).  prow0 = first packed row (= orig_row/2) of this K-chunk.
// ---------------------------------------------------------------------------
__device__ __forceinline__ void gemm2x4(const float* __restrict__ A0,
                                        const float* __restrict__ A1,
                                        const v2f* __restrict__ Wp,
                                        int prow0, int K, int cb,
                                        int mrow, int half,
                                        v8f acc[2][4]) {
    for (int kk = 0; kk < K; kk += 4) {
        const int kr = kk + 2 * half;
        const v2f a0 = *(const v2f*)(A0 + kr);
        const v2f a1 = *(const v2f*)(A1 + kr);
        const v2f* bp = Wp + (size_t)(prow0 + (kk >> 1) + half) * HDIM + cb + mrow;
#pragma unroll
        for (int c = 0; c < 4; ++c) {
            const v2f b = bp[c * 16];            // single b64 coalesced load
            acc[0][c] = wmma_f32(a0, b, acc[0][c]);
            acc[1][c] = wmma_f32(a1, b, acc[1][c]);
        }
    }
}

// 1 row-tile x 4 col-tiles (node kernel).
__device__ __forceinline__ void gemm1x4(const float* __restrict__ A0,
                                        const v2f* __restrict__ Wp,
                                        int prow0, int K, int cb,
                                        int mrow, int half,
                                        v8f acc[4]) {
    for (int kk = 0; kk < K; kk += 4) {
        const int kr = kk + 2 * half;
        const v2f a0 = *(const v2f*)(A0 + kr);
        const v2f* bp = Wp + (size_t)(prow0 + (kk >> 1) + half) * HDIM + cb + mrow;
#pragma unroll
        for (int c = 0; c < 4; ++c)
            acc[c] = wmma_f32(a0, bp[c * 16], acc[c]);
    }
}

// ---------------------------------------------------------------------------
// Weight pair-packing: Wp[p*128 + c] = {W[2p][c], W[2p+1][c]}.
// ---------------------------------------------------------------------------
__global__ void pack_pairs_kernel(const float* __restrict__ W,
                                  float* __restrict__ Wp, int npairs) {
    const int idx = blockIdx.x * blockDim.x + threadIdx.x;   // pair*128 + col
    if (idx >= npairs * HDIM) return;
    const int p = idx / HDIM, c = idx % HDIM;
    Wp[(size_t)idx * 2 + 0] = W[(size_t)(2 * p) * HDIM + c];
    Wp[(size_t)idx * 2 + 1] = W[(size_t)(2 * p + 1) * HDIM + c];
}

// ---------------------------------------------------------------------------
// Edge kernel: one wave (= one block) per 32-edge tile (2 x 16-row tiles).
// ---------------------------------------------------------------------------
__global__ __launch_bounds__(32)
void egnn_edge_kernel(const float* __restrict__ h,
                      const float* __restrict__ x,
                      const int*   __restrict__ eidx,   // [2,E] int32
                      const float* __restrict__ eattr,  // [E,16]
                      const float* __restrict__ We1,  const float* __restrict__ be1,
                      const float* __restrict__ be2,
                      const float* __restrict__ bx1,
                      const float* __restrict__ Wx2,  const float* __restrict__ bx2,
                      const float* __restrict__ We1p,   // packed (272 rows)
                      const float* __restrict__ We2p,   // packed (128 rows)
                      const float* __restrict__ Wx1p,   // packed (128 rows)
                      float* __restrict__ agg,          // [N,128] atomics
                      float* __restrict__ dxb)          // [N,3]   atomics
{
    __shared__ float t1[32 * LDS_LD];
    __shared__ float tm[32 * LDS_LD];
    __shared__ float wdist[32];
    __shared__ float wdir[32 * 3];
    __shared__ int   wsrc[32];
    __shared__ int   wdst[32];

    const int lane = threadIdx.x & 31;
    const int half = lane >> 4;
    const int mrow = lane & 15;
    const int tile = blockIdx.x;                  // 32 edges per tile
    const int ebase = tile * 32;

    // --- per-edge geometry: one lane per edge (all 32 lanes active) ---
    {
        const int e = ebase + lane;
        const int s = eidx[e];
        const int d = eidx[N_EDGES + e];
        wsrc[lane] = s;
        wdst[lane] = d;
        const float d0 = x[d * 3 + 0] - x[s * 3 + 0];
        const float d1 = x[d * 3 + 1] - x[s * 3 + 1];
        const float d2 = x[d * 3 + 2] - x[s * 3 + 2];
        const float dd = d0 * d0 + d1 * d1 + d2 * d2;
        wdist[lane] = sqrtf(dd + 1e-9f);              // sqrt(sum + eps)
        const float inv = 1.0f / (sqrtf(dd) + 1e-9f); // norm + eps
        wdir[lane * 3 + 0] = d0 * inv;
        wdir[lane * 3 + 1] = d1 * inv;
        wdir[lane * 3 + 2] = d2 * inv;
    }

    const v2f* We1v = (const v2f*)We1p;
    const v2f* We2v = (const v2f*)We2p;
    const v2f* Wx1v = (const v2f*)Wx1p;

    // A-row pointers for the two 16-row tiles (rows = edges ebase + t*16 + mrow).
    const float* hi0 = h + (size_t)wsrc[mrow] * HDIM;
    const float* hi1 = h + (size_t)wsrc[16 + mrow] * HDIM;
    const float* hj0 = h + (size_t)wdst[mrow] * HDIM;
    const float* hj1 = h + (size_t)wdst[16 + mrow] * HDIM;
    const float* ea0 = eattr + (size_t)(ebase + mrow) * EDIM;
    const float* ea1 = eattr + (size_t)(ebase + 16 + mrow) * EDIM;

    // ---- message layer 1: t1 = silu([h_i|h_j|attr|dist] @ We1 + be1) ----
    for (int cb = 0; cb < HDIM; cb += 64) {
        v8f acc[2][4];
#pragma unroll
        for (int t = 0; t < 2; ++t)
#pragma unroll
            for (int c = 0; c < 4; ++c) acc[t][c] = (v8f)0.f;
        gemm2x4(hi0, hi1, We1v, 0,   HDIM, cb, mrow, half, acc);  // rows 0..127
        gemm2x4(hj0, hj1, We1v, 64,  HDIM, cb, mrow, half, acc);  // rows 128..255
        gemm2x4(ea0, ea1, We1v, 128, EDIM, cb, mrow, half, acc);  // rows 256..271
#pragma unroll
        for (int c = 0; c < 4; ++c) {
            const int   col  = cb + c * 16 + mrow;
            const float bias = be1[col];
            const float w272 = We1[(size_t)272 * HDIM + col];     // dist row
#pragma unroll
            for (int t = 0; t < 2; ++t)
#pragma unroll
                for (int r = 0; r < 8; ++r) {
                    const int M = t * 16 + r + 8 * half;
                    const float v = acc[t][c][r] + bias + wdist[M] * w272;
                    t1[M * LDS_LD + col] = silu_f(v);
                }
        }
    }

    const float* t1r0 = t1 + mrow * LDS_LD;
    const float* t1r1 = t1 + (16 + mrow) * LDS_LD;
    const float* tmr0 = tm + mrow * LDS_LD;
    const float* tmr1 = tm + (16 + mrow) * LDS_LD;

    // ---- message layer 2: m = t1 @ We2 + be2 ; scatter into agg ----
    for (int cb = 0; cb < HDIM; cb += 64) {
        v8f acc[2][4];
#pragma unroll
        for (int t = 0; t < 2; ++t)
#pragma unroll
            for (int c = 0; c < 4; ++c) acc[t][c] = (v8f)0.f;
        gemm2x4(t1r0, t1r1, We2v, 0, HDIM, cb, mrow, half, acc);
#pragma unroll
        for (int c = 0; c < 4; ++c) {
            const int   col  = cb + c * 16 + mrow;
            const float bias = be2[col];
#pragma unroll
            for (int t = 0; t < 2; ++t)
#pragma unroll
                for (int r = 0; r < 8; ++r) {
                    const int M = t * 16 + r + 8 * half;
                    const float v = acc[t][c][r] + bias;
                    tm[M * LDS_LD + col] = v;
                    atomicAdd(&agg[(size_t)wdst[M] * HDIM + col], v);
                }
        }
    }

    // ---- gate layer 1: t1 = silu(m @ Wx1 + bx1)  (reuse t1) ----
    for (int cb = 0; cb < HDIM; cb += 64) {
        v8f acc[2][4];
#pragma unroll
        for (int t = 0; t < 2; ++t)
#pragma unroll
            for (int c = 0; c < 4; ++c) acc[t][c] = (v8f)0.f;
        gemm2x4(tmr0, tmr1, Wx1v, 0, HDIM, cb, mrow, half, acc);
#pragma unroll
        for (int c = 0; c < 4; ++c) {
            const int   col  = cb + c * 16 + mrow;
            const float bias = bx1[col];
#pragma unroll
            for (int t = 0; t < 2; ++t)
#pragma unroll
                for (int r = 0; r < 8; ++r) {
                    const int M = t * 16 + r + 8 * half;
                    t1[M * LDS_LD + col] = silu_f(acc[t][c][r] + bias);
                }
        }
    }

    // ---- gate = t1 . Wx2 + bx2 ; dx scatter (one lane per edge) ----
    {
        float g = bx2[0];
        const float* tr = t1 + lane * LDS_LD;
        for (int c = 0; c < HDIM; ++c) g += tr[c] * Wx2[c];
        const int d = wdst[lane];
        atomicAdd(&dxb[(size_t)d * 3 + 0], wdir[lane * 3 + 0] * g);
        atomicAdd(&dxb[(size_t)d * 3 + 1], wdir[lane * 3 + 1] * g);
        atomicAdd(&dxb[(size_t)d * 3 + 2], wdir[lane * 3 + 2] * g);
    }
}

// ---------------------------------------------------------------------------
// Node kernel: one wave per 16-node tile, 2 waves per block.
// ---------------------------------------------------------------------------
__global__ __launch_bounds__(64)
void egnn_node_kernel(const float* __restrict__ h,
                      const float* __restrict__ x,
                      const float* __restrict__ agg,
                      const float* __restrict__ dxb,
                      const float* __restrict__ bh1,
                      const float* __restrict__ bh2,
                      const float* __restrict__ ln_g, const float* __restrict__ ln_b,
                      const float* __restrict__ Wh1p,  // packed (256 rows)
                      const float* __restrict__ Wh2p,  // packed (128 rows)
                      float* __restrict__ hout,        // [N,128]
                      float* __restrict__ xout)        // [N,3]
{
    __shared__ float smem[2 * 2 * 16 * LDS_LD];

    const int wave = threadIdx.x >> 5;
    const int lane = threadIdx.x & 31;
    const int half = lane >> 4;
    const int mrow = lane & 15;
    const int tile = blockIdx.x * 2 + wave;
    if (tile >= N_NODES / 16) return;

    float* tt = smem + wave * (2 * 16 * LDS_LD);
    float* ty = tt + 16 * LDS_LD;

    const v2f* Wh1v = (const v2f*)Wh1p;
    const v2f* Wh2v = (const v2f*)Wh2p;

    const int    nrow  = tile * 16 + mrow;
    const float* h_row = h   + (size_t)nrow * HDIM;
    const float* a_row = agg + (size_t)nrow * HDIM;

    // ---- tt = silu([h|agg] @ Wh1 + bh1) ----
    for (int cb = 0; cb < HDIM; cb += 64) {
        v8f acc[4];
#pragma unroll
        for (int c = 0; c < 4; ++c) acc[c] = (v8f)0.f;
        gemm1x4(h_row, Wh1v, 0,  HDIM, cb, mrow, half, acc);   // rows 0..127
        gemm1x4(a_row, Wh1v, 64, HDIM, cb, mrow, half, acc);   // rows 128..255
#pragma unroll
        for (int c = 0; c < 4; ++c) {
            const int   col  = cb + c * 16 + mrow;
            const float bias = bh1[col];
#pragma unroll
            for (int r = 0; r < 8; ++r) {
                const int M = r + 8 * half;
                tt[M * LDS_LD + col] = silu_f(acc[c][r] + bias);
            }
        }
    }

    // ---- y = h + (tt @ Wh2 + bh2) ----
    for (int cb = 0; cb < HDIM; cb += 64) {
        v8f acc[4];
#pragma unroll
        for (int c = 0; c < 4; ++c) acc[c] = (v8f)0.f;
        gemm1x4(tt + mrow * LDS_LD, Wh2v, 0, HDIM, cb, mrow, half, acc);
#pragma unroll
        for (int c = 0; c < 4; ++c) {
            const int   col  = cb + c * 16 + mrow;
            const float bias = bh2[col];
#pragma unroll
            for (int r = 0; r < 8; ++r) {
                const int M = r + 8 * half;
                const int row = tile * 16 + M;
                ty[M * LDS_LD + col] = h[(size_t)row * HDIM + col] + acc[c][r] + bias;
            }
        }
    }

    // ---- per-row LayerNorm + x update (one lane per node) ----
    if (half == 0) {
        const float* yr = ty + mrow * LDS_LD;
        float mu = 0.f;
        for (int c = 0; c < HDIM; ++c) mu += yr[c];
        mu *= (1.0f / HDIM);
        float var = 0.f;
        for (int c = 0; c < HDIM; ++c) { const float t = yr[c] - mu; var += t * t; }
        var *= (1.0f / HDIM);
        const float is = rsqrtf(var + 1e-5f);
        const size_t ro = (size_t)nrow * HDIM;
        for (int c = 0; c < HDIM; ++c)
            hout[ro + c] = (yr[c] - mu) * is * ln_g[c] + ln_b[c];
        xout[(size_t)nrow * 3 + 0] = x[(size_t)nrow * 3 + 0] + dxb[(size_t)nrow * 3 + 0];
        xout[(size_t)nrow * 3 + 1] = x[(size_t)nrow * 3 + 1] + dxb[(size_t)nrow * 3 + 1];
        xout[(size_t)nrow * 3 + 2] = x[(size_t)nrow * 3 + 2] + dxb[(size_t)nrow * 3 + 2];
    }
}

// ---------------------------------------------------------------------------
extern "C" void kernel_launch(void* const* d_in, const int* in_sizes, int n_in,
                              void* d_out, int out_size, void* d_ws, size_t ws_size,
                              hipStream_t stream) {
    const float* h     = (const float*)d_in[0];
    const float* x     = (const float*)d_in[1];
    const int*   eidx  = (const int*)d_in[2];   // jnp.int64 coerced to int32 (x64 off)
    const float* eattr = (const float*)d_in[3];
    const float* We1 = (const float*)d_in[4];  const float* be1 = (const float*)d_in[5];
    const float* We2 = (const float*)d_in[6];  const float* be2 = (const float*)d_in[7];
    const float* Wh1 = (const float*)d_in[8];  const float* bh1 = (const float*)d_in[9];
    const float* Wh2 = (const float*)d_in[10]; const float* bh2 = (const float*)d_in[11];
    const float* Wx1 = (const float*)d_in[12]; const float* bx1 = (const float*)d_in[13];
    const float* Wx2 = (const float*)d_in[14]; const float* bx2 = (const float*)d_in[15];
    const float* ln_g = (const float*)d_in[16]; const float* ln_b = (const float*)d_in[17];

    // Workspace layout (floats).
    float* agg  = (float*)d_ws;                      // [N,128]
    float* dxb  = agg  + (size_t)N_NODES * HDIM;     // [N,3]
    float* We1p = dxb  + (size_t)N_NODES * 3;        // 272*128
    float* We2p = We1p + 272 * HDIM;                 // 128*128
    float* Wx1p = We2p + 128 * HDIM;                 // 128*128
    float* Wh1p = Wx1p + 128 * HDIM;                 // 256*128
    float* Wh2p = Wh1p + 256 * HDIM;                 // 128*128

    hipMemsetAsync(d_ws, 0,
                   ((size_t)N_NODES * HDIM + (size_t)N_NODES * 3) * sizeof(float),
                   stream);

    // Pair-pack the GEMM weights (one-shot, tiny).
    const int pk1 = 136 * HDIM, pk2 = 64 * HDIM, pk3 = 128 * HDIM;
    pack_pairs_kernel<<<(pk1 + 255) / 256, 256, 0, stream>>>(We1, We1p, 136);
    pack_pairs_kernel<<<(pk2 + 255) / 256, 256, 0, stream>>>(We2, We2p, 64);
    pack_pairs_kernel<<<(pk2 + 255) / 256, 256, 0, stream>>>(Wx1, Wx1p, 64);
    pack_pairs_kernel<<<(pk3 + 255) / 256, 256, 0, stream>>>(Wh1, Wh1p, 128);
    pack_pairs_kernel<<<(pk2 + 255) / 256, 256, 0, stream>>>(Wh2, Wh2p, 64);
    (void)pk1; (void)pk3;

    float* hout = (float*)d_out;
    float* xout = hout + (size_t)N_NODES * HDIM;

    egnn_edge_kernel<<<N_EDGES / 32, 32, 0, stream>>>(
        h, x, eidx, eattr, We1, be1, be2, bx1, Wx2, bx2,
        We1p, We2p, Wx1p, agg, dxb);

    const int ntiles = N_NODES / 16;                 // 3125
    egnn_node_kernel<<<(ntiles + 1) / 2, 64, 0, stream>>>(
        h, x, agg, dxb, bh1, bh2, ln_g, ln_b, Wh1p, Wh2p, hout, xout);
}